// MagLinear_50414326120961
// MI455X (gfx1250) — compile-verified
//
#include <hip/hip_runtime.h>
#include <hip/hip_bf16.h>
#include <math.h>

typedef __attribute__((ext_vector_type(2))) float v2f;
typedef __attribute__((ext_vector_type(4))) float v4f;
typedef __attribute__((ext_vector_type(8))) float v8f;

#define D_DIM 512
#define B_DIM 512
#define BN_EPS 2e-5f
#define NORM_EPS 1e-12f
#define SCALE_S 64.0f
#define L_MARGIN 0.45f
#define U_MARGIN 0.8f

#define NTILE 64
#define KC 32
#define KCPAD 34
#define NCHUNK (D_DIM / KC)   // 16

// ---------------------------------------------------------------------------
// Kernel 1: BatchNorm statistics -> per-feature scale/shift
// ---------------------------------------------------------------------------
__global__ void bn_stats_kernel(const float* __restrict__ x,
                                const float* __restrict__ gamma,
                                const float* __restrict__ beta,
                                float* __restrict__ scale,
                                float* __restrict__ shift) {
  int d = blockIdx.x * blockDim.x + threadIdx.x;
  if (d >= D_DIM) return;
  float s = 0.f, ss = 0.f;
  for (int b = 0; b < B_DIM; ++b) {
    float v = x[(size_t)b * D_DIM + d];
    s += v;
    ss += v * v;
  }
  float mean = s * (1.0f / (float)B_DIM);
  float var = ss * (1.0f / (float)B_DIM) - mean * mean;
  float sc = gamma[d] / sqrtf(var + BN_EPS);
  scale[d] = sc;
  shift[d] = beta[d] - mean * sc;
}

// ---------------------------------------------------------------------------
// Kernel 2: per-row BN apply, row L2 norm, margin cos/sin, x_unit
// one block (256 threads) per row; D=512 -> 2 elements/thread
// ---------------------------------------------------------------------------
__global__ void rownorm_kernel(const float* __restrict__ x,
                               const float* __restrict__ scale,
                               const float* __restrict__ shift,
                               const int* __restrict__ la_p,
                               const int* __restrict__ ua_p,
                               float* __restrict__ xunit,
                               float* __restrict__ cosm,
                               float* __restrict__ sinm,
                               float* __restrict__ xnorm_out) {
  const int row = blockIdx.x;
  const int tid = threadIdx.x;      // 0..255
  const int lane = tid & 31, wid = tid >> 5;
  const size_t base = (size_t)row * D_DIM;
  float e0 = x[base + tid] * scale[tid] + shift[tid];
  float e1 = x[base + tid + 256] * scale[tid + 256] + shift[tid + 256];
  float ss = e0 * e0 + e1 * e1;
#pragma unroll
  for (int off = 16; off >= 1; off >>= 1)
    ss += __shfl_down(ss, off, 32);   // wave32 reduction
  __shared__ float wsum[8];
  __shared__ float s_inv;
  if (lane == 0) wsum[wid] = ss;
  __syncthreads();
  if (tid == 0) {
    float t = 0.f;
#pragma unroll
    for (int i = 0; i < 8; ++i) t += wsum[i];
    float nrm = sqrtf(t);
    float la = (float)la_p[0], ua = (float)ua_p[0];
    float xn = fminf(fmaxf(nrm, la), ua);
    float ada = (U_MARGIN - L_MARGIN) / (ua - la) * (xn - la) + L_MARGIN;
    cosm[row] = cosf(ada);
    sinm[row] = sinf(ada);
    xnorm_out[row] = xn;
    s_inv = 1.0f / fmaxf(nrm, NORM_EPS);
  }
  __syncthreads();
  float inv = s_inv;
  xunit[base + tid] = e0 * inv;
  xunit[base + tid + 256] = e1 * inv;
}

// ---------------------------------------------------------------------------
// Kernel 3: per-column inverse L2 norm of weight
// ---------------------------------------------------------------------------
__global__ void wnorm_kernel(const float* __restrict__ w, int Nw,
                             float* __restrict__ invwn) {
  int n = blockIdx.x * blockDim.x + threadIdx.x;
  if (n >= Nw) return;
  float ss = 0.f;
#pragma unroll 4
  for (int d = 0; d < D_DIM; ++d) {
    float v = w[(size_t)d * Nw + n];
    ss += v * v;
  }
  invwn[n] = 1.0f / fmaxf(sqrtf(ss), NORM_EPS);
}

// ---------------------------------------------------------------------------
// Kernel 4: fp32 WMMA GEMM (M=512 full per block, NTILE=64 per block)
//   C = x_unit @ weight, normalized by invwn in epilogue + MagFace margin
// ---------------------------------------------------------------------------
__device__ __forceinline__ void load_w_chunk(const float* __restrict__ W, int Nw,
                                             int n0, int j4, int drow, int k0,
                                             v4f pf[2]) {
#pragma unroll
  for (int p = 0; p < 2; ++p) {
    int d = k0 + drow + p * 16;
    int n = n0 + j4;
    if (n + 3 < Nw) {
      pf[p] = __builtin_nontemporal_load((const v4f*)(W + (size_t)d * Nw + n));
    } else {
      v4f t;
#pragma unroll
      for (int i = 0; i < 4; ++i)
        t[i] = (n + i < Nw) ? W[(size_t)d * Nw + n + i] : 0.0f;
      pf[p] = t;
    }
  }
}

__device__ __forceinline__ void store_w_lds(float Bt[NTILE][KCPAD],
                                            int j4, int drow, const v4f pf[2]) {
#pragma unroll
  for (int p = 0; p < 2; ++p)
#pragma unroll
    for (int i = 0; i < 4; ++i)
      Bt[j4 + i][drow + p * 16] = pf[p][i];   // transposed: [n][k]
}

__global__ __launch_bounds__(256, 1) void magface_gemm_kernel(
    const float* __restrict__ A,      // x_unit [512][512]
    const float* __restrict__ W,      // weight [512][Nw]
    const float* __restrict__ invwn,  // [Nw]
    const float* __restrict__ cosm,   // [512]
    const float* __restrict__ sinm,   // [512]
    float* __restrict__ out0, float* __restrict__ out1, int Nw) {
  __shared__ float Bt[2][NTILE][KCPAD];   // 17.4 KB, double buffered
  const int tid = threadIdx.x;
  const int wave = tid >> 5;
  const int lane = tid & 31;
  const int n0 = blockIdx.x * NTILE;
  const int mbase = wave * 64;            // 8 waves cover M=512
  const int j4 = (tid & 15) * 4;          // coop-load column group
  const int drow = tid >> 4;              // coop-load k row 0..15
  const int mlane = lane & 15;            // fragment row/col within tile
  const int kk = (lane >> 4) * 2;         // fp32 WMMA k-pair select
  const int rbase = (lane >> 4) * 8;      // C/D row base per half-wave

  v8f acc[4][4];
#pragma unroll
  for (int mt = 0; mt < 4; ++mt)
#pragma unroll
    for (int nt = 0; nt < 4; ++nt)
#pragma unroll
      for (int r = 0; r < 8; ++r) acc[mt][nt][r] = 0.0f;

  v4f pf[2];
  load_w_chunk(W, Nw, n0, j4, drow, 0, pf);
  store_w_lds(Bt[0], j4, drow, pf);
  __syncthreads();

  int cur = 0;
  for (int kc = 0; kc < NCHUNK; ++kc) {
    if (kc + 1 < NCHUNK)
      load_w_chunk(W, Nw, n0, j4, drow, (kc + 1) * KC, pf);   // prefetch next

    const float* Arow = A + (size_t)(mbase + mlane) * D_DIM + kc * KC + kk;
#pragma unroll
    for (int ks = 0; ks < KC / 4; ++ks) {
      v2f a[4], b[4];
#pragma unroll
      for (int mt = 0; mt < 4; ++mt)
        a[mt] = *(const v2f*)(Arow + mt * 16 * D_DIM + ks * 4);
#pragma unroll
      for (int nt = 0; nt < 4; ++nt)
        b[nt] = *(const v2f*)(&Bt[cur][nt * 16 + mlane][ks * 4 + kk]);
#pragma unroll
      for (int mt = 0; mt < 4; ++mt)
#pragma unroll
        for (int nt = 0; nt < 4; ++nt)
          acc[mt][nt] = __builtin_amdgcn_wmma_f32_16x16x4_f32(
              false, a[mt], false, b[nt], (short)0, acc[mt][nt], false, false);
    }
    __syncthreads();
    if (kc + 1 < NCHUNK)
      store_w_lds(Bt[cur ^ 1], j4, drow, pf);
    __syncthreads();
    cur ^= 1;
  }

  // Epilogue: normalize, clamp, margin trig, scale, store both outputs
#pragma unroll
  for (int nt = 0; nt < 4; ++nt) {
    int n = n0 + nt * 16 + mlane;
    if (n < Nw) {
      float iw = invwn[n];
#pragma unroll
      for (int mt = 0; mt < 4; ++mt) {
        int m0 = mbase + mt * 16 + rbase;
#pragma unroll
        for (int r = 0; r < 8; ++r) {
          int m = m0 + r;
          float cosv = fminf(fmaxf(acc[mt][nt][r] * iw, -1.0f), 1.0f);
          float sinv = sqrtf(fmaxf(1.0f - cosv * cosv, 0.0f));
          float ctm = cosv * cosm[m] - sinv * sinm[m];
          ctm = (cosv > 0.0f) ? ctm : cosv;
          size_t idx = (size_t)m * (size_t)Nw + (size_t)n;
          __builtin_nontemporal_store(SCALE_S * cosv, out0 + idx);
          __builtin_nontemporal_store(SCALE_S * ctm, out1 + idx);
        }
      }
    }
  }
}

// ---------------------------------------------------------------------------
extern "C" void kernel_launch(void* const* d_in, const int* in_sizes, int n_in,
                              void* d_out, int out_size, void* d_ws, size_t ws_size,
                              hipStream_t stream) {
  const float* x = (const float*)d_in[0];       // [512,512]
  const float* weight = (const float*)d_in[1];  // [512,N]
  const float* gamma = (const float*)d_in[2];   // [512]
  const float* beta = (const float*)d_in[3];    // [512]
  const int* la = (const int*)d_in[4];          // scalar
  const int* ua = (const int*)d_in[5];          // scalar
  const int Nw = in_sizes[1] / D_DIM;           // 100000

  float* ws = (float*)d_ws;
  float* w_scale = ws;                          // 512
  float* w_shift = ws + 512;                    // 512
  float* w_cosm = ws + 1024;                    // 512
  float* w_sinm = ws + 1536;                    // 512
  float* w_xunit = ws + 2048;                   // 512*512
  float* w_invwn = ws + 2048 + B_DIM * D_DIM;   // Nw

  float* out = (float*)d_out;
  float* out0 = out;                                  // 64*cos_theta
  float* out1 = out + (size_t)B_DIM * (size_t)Nw;     // 64*cos_theta_m
  float* xnorm = out + 2 * (size_t)B_DIM * (size_t)Nw;  // x_norm [512]

  hipLaunchKernelGGL(bn_stats_kernel, dim3(2), dim3(256), 0, stream,
                     x, gamma, beta, w_scale, w_shift);
  hipLaunchKernelGGL(rownorm_kernel, dim3(B_DIM), dim3(256), 0, stream,
                     x, w_scale, w_shift, la, ua, w_xunit, w_cosm, w_sinm, xnorm);
  hipLaunchKernelGGL(wnorm_kernel, dim3((Nw + 255) / 256), dim3(256), 0, stream,
                     weight, Nw, w_invwn);
  hipLaunchKernelGGL(magface_gemm_kernel, dim3((Nw + NTILE - 1) / NTILE),
                     dim3(256), 0, stream,
                     w_xunit, weight, w_invwn, w_cosm, w_sinm, out0, out1, Nw);
}